// EBM_74792560492767
// MI455X (gfx1250) — compile-verified
//
#include <hip/hip_runtime.h>

typedef float v2f __attribute__((ext_vector_type(2)));
typedef float v8f __attribute__((ext_vector_type(8)));

#define C_DIM  768
#define V_DIM  50257
#define BT_DIM 2048

// ---------------------------------------------------------------------------
// GEMM: out[m][v] = -sum_k h[m][k] * W[v][k] via V_WMMA_F32_16X16X4_F32.
// Wave tile = 128(M) x 16(N): 8 f32 accumulators, 8 A b64-loads + 1 B b64-load
// per k-step, 8 WMMAs. 8 waves/WG -> 128 x 128 output per workgroup.
// This halves total W (147MB) L2 re-reads vs a 64-row block: 16 passes, not 32.
// A (16x4 f32): lanes 0-15 M=0..15 K={k,k+1}; lanes 16-31 M=0..15 K={k+2,k+3}.
// B (4x16 f32): same pairing with N on lanes. One float2 per lane per operand.
// ---------------------------------------------------------------------------
__global__ __launch_bounds__(256) void ebm_gemm_wmma(
    const float* __restrict__ h, const float* __restrict__ W,
    float* __restrict__ out)
{
    const int lane = threadIdx.x & 31;
    const int wave = threadIdx.x >> 5;          // 0..7 -> N sub-tile
    const int half = lane >> 4;                 // 0/1 -> K pair select
    const int lm   = lane & 15;                 // M (for A) / N (for B)

    const int col  = blockIdx.x * 128 + wave * 16 + lm;   // B row / D column
    const int colc = (col < V_DIM) ? col : (V_DIM - 1);   // clamp for safe loads
    const int row0 = blockIdx.y * 128;                    // 8 M sub-tiles of 16

    const float* brow = W + (size_t)colc * C_DIM + 2 * half;
    const float* arow = h + (size_t)(row0 + lm) * C_DIM + 2 * half;

    v8f acc[8];
    #pragma unroll
    for (int mt = 0; mt < 8; ++mt) acc[mt] = (v8f){};

    #pragma unroll 2
    for (int k = 0; k < C_DIM; k += 4) {
        const v2f b = *(const v2f*)(brow + k);
        #pragma unroll
        for (int mt = 0; mt < 8; ++mt) {
            const v2f a = *(const v2f*)(arow + (size_t)mt * 16 * C_DIM + k);
            acc[mt] = __builtin_amdgcn_wmma_f32_16x16x4_f32(
                false, a, false, b, (short)0, acc[mt], false, false);
        }
    }

    if (col < V_DIM) {
        // D layout: VGPR r -> row (r + 8*half), lane%16 -> column
        #pragma unroll
        for (int mt = 0; mt < 8; ++mt) {
            #pragma unroll
            for (int r = 0; r < 8; ++r) {
                const int m = row0 + mt * 16 + r + 8 * half;
                out[(size_t)m * V_DIM + col] = -acc[mt][r];
            }
        }
    }
}

// ---------------------------------------------------------------------------
// Fused refinement, register-resident rows.
// One 1024-thread WG per row; each thread owns 50 row elements in VGPRs
// (49 full strides + 81-lane tail). No big LDS allocation -> no 1-WG/WGP cap.
// Block reductions: wave32 shfl_xor butterfly + 32-entry LDS scratch
// (3 barriers per reduction instead of a 10-level tree).
// ---------------------------------------------------------------------------
#define NFULL 49                       // V_DIM / 1024
#define NTAIL (V_DIM - NFULL * 1024)   // 81

__device__ __forceinline__ float wave_red_sum(float v) {
    #pragma unroll
    for (int o = 16; o > 0; o >>= 1) v += __shfl_xor(v, o, 32);
    return v;
}
__device__ __forceinline__ float wave_red_max(float v) {
    #pragma unroll
    for (int o = 16; o > 0; o >>= 1) v = fmaxf(v, __shfl_xor(v, o, 32));
    return v;
}

__device__ __forceinline__ float block_max(float v, float* s) {
    const int wid = threadIdx.x >> 5, lane = threadIdx.x & 31;
    v = wave_red_max(v);
    if (lane == 0) s[wid] = v;
    __syncthreads();
    if (wid == 0) {
        float x = wave_red_max(s[lane]);
        if (lane == 0) s[0] = x;
    }
    __syncthreads();
    float r = s[0];
    __syncthreads();
    return r;
}

__device__ __forceinline__ float block_sum(float v, float* s) {
    const int wid = threadIdx.x >> 5, lane = threadIdx.x & 31;
    v = wave_red_sum(v);
    if (lane == 0) s[wid] = v;
    __syncthreads();
    if (wid == 0) {
        float x = wave_red_sum(s[lane]);
        if (lane == 0) s[0] = x;
    }
    __syncthreads();
    float r = s[0];
    __syncthreads();
    return r;
}

__device__ __forceinline__ void block_sum2(float a, float b, float* s0, float* s1,
                                           float& ra, float& rb) {
    const int wid = threadIdx.x >> 5, lane = threadIdx.x & 31;
    a = wave_red_sum(a);
    b = wave_red_sum(b);
    if (lane == 0) { s0[wid] = a; s1[wid] = b; }
    __syncthreads();
    if (wid == 0) {
        float x = wave_red_sum(s0[lane]);
        float y = wave_red_sum(s1[lane]);
        if (lane == 0) { s0[0] = x; s1[0] = y; }
    }
    __syncthreads();
    ra = s0[0]; rb = s1[0];
    __syncthreads();
}

__global__ __launch_bounds__(1024) void ebm_refine(
    float* __restrict__ logits, const float* __restrict__ alpha_p,
    const int* __restrict__ steps_p)
{
    __shared__ float s0[32], s1[32];

    const int  tid      = threadIdx.x;
    const bool has_tail = (tid < NTAIL);
    float* g = logits + (size_t)blockIdx.x * V_DIM;

    float r[NFULL + 1];
    #pragma unroll
    for (int j = 0; j < NFULL; ++j) r[j] = g[tid + j * 1024];
    r[NFULL] = has_tail ? g[tid + NFULL * 1024] : 0.0f;

    const float alpha    = *alpha_p;
    const int   steps    = *steps_p;
    const float invdenom = 1.0f / (float)BT_DIM;
    const float invV     = 1.0f / (float)V_DIM;

    for (int s = 0; s < steps; ++s) {
        // 1) row max
        float m = -3.402823466e38f;
        #pragma unroll
        for (int j = 0; j < NFULL; ++j) m = fmaxf(m, r[j]);
        if (has_tail) m = fmaxf(m, r[NFULL]);
        m = block_max(m, s0);

        // 2) sum(exp) and sum(exp*l) in one pass
        float se = 0.0f, sl = 0.0f;
        #pragma unroll
        for (int j = 0; j < NFULL; ++j) {
            const float e = __expf(r[j] - m);
            se += e; sl += e * r[j];
        }
        if (has_tail) {
            const float e = __expf(r[NFULL] - m);
            se += e; sl += e * r[NFULL];
        }
        float Z, SL;
        block_sum2(se, sl, s0, s1, Z, SL);
        const float invZ = 1.0f / Z;
        const float E    = SL * invZ;          // E_p[logits]

        // 3) analytic-grad update + accumulate new-row sum
        float ls = 0.0f;
        #pragma unroll
        for (int j = 0; j < NFULL; ++j) {
            float l = r[j];
            const float p = __expf(l - m) * invZ;
            float grd = -(p * (1.0f + l - E)) * invdenom;
            grd = fminf(fmaxf(grd, -5.0f), 5.0f);
            l -= alpha * grd;
            r[j] = l; ls += l;
        }
        if (has_tail) {
            float l = r[NFULL];
            const float p = __expf(l - m) * invZ;
            float grd = -(p * (1.0f + l - E)) * invdenom;
            grd = fminf(fmaxf(grd, -5.0f), 5.0f);
            l -= alpha * grd;
            r[NFULL] = l; ls += l;
        }
        const float mean = block_sum(ls, s0) * invV;

        // 4) mean-center
        #pragma unroll
        for (int j = 0; j < NFULL; ++j) r[j] -= mean;
        if (has_tail) r[NFULL] -= mean;
    }

    #pragma unroll
    for (int j = 0; j < NFULL; ++j) g[tid + j * 1024] = r[j];
    if (has_tail) g[tid + NFULL * 1024] = r[NFULL];
}

// ---------------------------------------------------------------------------
extern "C" void kernel_launch(void* const* d_in, const int* in_sizes, int n_in,
                              void* d_out, int out_size, void* d_ws, size_t ws_size,
                              hipStream_t stream) {
    const float* h     = (const float*)d_in[0];   // (2,1024,768) f32
    const float* W     = (const float*)d_in[1];   // (50257,768)  f32
    const float* alpha = (const float*)d_in[2];   // scalar f32
    const int*   steps = (const int*)d_in[3];     // scalar int
    float* out = (float*)d_out;                   // (2,1024,50257) f32

    dim3 gemm_grid((V_DIM + 127) / 128, BT_DIM / 128);
    ebm_gemm_wmma<<<gemm_grid, 256, 0, stream>>>(h, W, out);

    ebm_refine<<<BT_DIM, 1024, 0, stream>>>(out, alpha, steps);
}